// Model_48696339202401
// MI455X (gfx1250) — compile-verified
//
#include <hip/hip_runtime.h>
#include <math.h>

// MI455X / gfx1250, wave32. Sim core: V_WMMA_F32_16X16X4_F32 (fp32 precision =
// reference) over an LDS-resident slice of the signed tension matrix G.
// CDNA5 data movement: TENSOR_LOAD_TO_LDS stages each WG's 64KB G slice
// (with TDM LDS padding -> pitch 516), GLOBAL_LOAD_ASYNC_TO_LDS_B32 stages the
// 6KB shared state u each step (ASYNCcnt). 16 WGs run the 2048 sequential
// frames in lock-step via an atomic release/acquire grid barrier.

typedef float v2f __attribute__((ext_vector_type(2)));
typedef float v8f __attribute__((ext_vector_type(8)));
typedef unsigned u32x4 __attribute__((ext_vector_type(4)));
typedef int i32x4 __attribute__((ext_vector_type(4)));
typedef int i32x8 __attribute__((ext_vector_type(8)));

#define N_NODES   512
#define N_FRAMES  2048
#define NWG       16
#define ROWS_WG   32
#define TPB       256          // 8 waves of 32
#define GPITCH    516          // 512 data + 2-DW pad every 256 DW (TDM-compatible)
#define N_COEF    1025
#define N_SAMP    524288
#define TWO_PI    6.28318530717958647692f

// workspace layout (float offsets)
#define OFF_U0    0            // 1536
#define OFF_U1    1536         // 1536
#define OFF_CS    3072         // 512
#define OFF_IM    3584         // 512
#define OFF_CM    4096         // 512
#define OFF_MM    4608         // 512
#define OFF_CTR   5120         // 64
#define OFF_RECP  5184         // 2048*16
#define OFF_REC   37952        // 2048
#define OFF_XR    40000        // 1040
#define OFF_XI    41040        // 1040
#define WS_BASE_FLOATS 42112
#define OFF_G     42240        // 512*512 (TDM source copy of G)
#define WS_TDM_FLOATS (OFF_G + N_NODES * N_NODES)

// builtin takes address-space-typed int* (HIP prints AS1 as "__device__ int*")
#define ASP_GLOBAL(p) ((__attribute__((address_space(1))) int*)(unsigned long long)(p))
#define ASP_LDS(p)    ((__attribute__((address_space(3))) int*)(unsigned)(unsigned long long)(p))

#if __has_builtin(__builtin_amdgcn_global_load_async_to_lds_b32) && \
    __has_builtin(__builtin_amdgcn_s_wait_asynccnt)
#define HAVE_ASYNC_LDS 1
#else
#define HAVE_ASYNC_LDS 0
#endif

#if __has_builtin(__builtin_amdgcn_tensor_load_to_lds) && \
    __has_builtin(__builtin_amdgcn_s_wait_tensorcnt)
#define HAVE_TDM 1
#else
#define HAVE_TDM 0
#endif

__device__ __forceinline__ float zval(const float* tensions, int i, int k) {
  int lo = i < k ? i : k, hi = i < k ? k : i;
  return 4.0f * fabsf(tensions[lo * N_NODES + hi]) + 1e-8f;
}

// ---------------- prep: per-node constants, softmax, zero state -------------
__global__ void prep_kernel(const float* __restrict__ masses,
                            const float* __restrict__ tensions,
                            const float* __restrict__ mixer,
                            const float* __restrict__ cmask,
                            float* __restrict__ ws) {
  __shared__ float red[N_NODES];
  const int k = threadIdx.x;
  float m_eff = fabsf(masses[k]) * 10000.0f + 1e-8f;

  float cs = 0.0f;
  for (int i = 0; i < N_NODES; ++i) {
    if (i == k) continue;
    float z = zval(tensions, i, k);
    cs += (i < k) ? z : -z;            // cs[k] = sum_i G[k,i]
  }

  float mk = mixer[k];
  red[k] = mk; __syncthreads();
  for (int s = 256; s > 0; s >>= 1) { if (k < s) red[k] = fmaxf(red[k], red[k + s]); __syncthreads(); }
  float mx = red[0]; __syncthreads();
  float e = expf(mk - mx);
  red[k] = e; __syncthreads();
  for (int s = 256; s > 0; s >>= 1) { if (k < s) red[k] += red[k + s]; __syncthreads(); }
  float mix = e / red[0];

  ws[OFF_CS + k] = cs;
  ws[OFF_IM + k] = 1.0f / m_eff;
  ws[OFF_CM + k] = (cmask[k] > 0.0f) ? 1.0f : 0.0f;
  ws[OFF_MM + k] = mix * m_eff;
  ws[OFF_U0 + 3 * k + 0] = 0.0f;       // u = pos - nodes starts at 0
  ws[OFF_U0 + 3 * k + 1] = 0.0f;
  ws[OFF_U0 + 3 * k + 2] = 0.0f;
  if (k == 0) *(unsigned*)(ws + OFF_CTR) = 0u;
}

// ---------------- prep_g: materialize G row-major for the TDM ---------------
__global__ void prep_g(const float* __restrict__ tensions, float* __restrict__ ws) {
  int idx = blockIdx.x * TPB + threadIdx.x;          // 0 .. 512*512-1
  int k = idx >> 9, i = idx & 511;
  float v = 0.0f;
  if (i != k) { float z = zval(tensions, i, k); v = (i < k) ? z : -z; }
  ws[OFF_G + idx] = v;
}

// one fp32 WMMA K-step: D += A(16x4) * B(4x16)
__device__ __forceinline__ v8f wmma_step(const float* __restrict__ grow,
                                         const float* __restrict__ u_lds,
                                         int k0, int hi, int bidx, v8f c) {
  const int ka = k0 + 2 * hi;                    // A: K = 2*hi + vgpr
  v2f A = *(const v2f*)(grow + ka);              // ds_load_b64
  v2f B;
  B.x = u_lds[ka * 4 + bidx];                    // B: row K striped over lanes
  B.y = u_lds[(ka + 1) * 4 + bidx];
  return __builtin_amdgcn_wmma_f32_16x16x4_f32(false, A, false, B,
                                               (short)0, c, false, false);
}

// ---------------- sim: 2048 sequential frames, 16 WGs in lock-step ----------
__global__ void sim_kernel(const float* __restrict__ tensions,
                           const float* __restrict__ forces,
                           float* __restrict__ ws, int use_tdm) {
  __shared__ __align__(16) float Gs[ROWS_WG * GPITCH];
  __shared__ __align__(16) float u_lds[N_NODES * 4];     // col 3 kept zero
  __shared__ v8f  part[8 * 32];
  __shared__ float recs[4];

  const int wg   = blockIdx.x;
  const int tid  = threadIdx.x;
  const int lane = tid & 31;
  const int wave = tid >> 5;
  const int rowbase = wg * ROWS_WG;

  float* u0 = ws + OFF_U0;
  float* u1 = ws + OFF_U1;
  unsigned* counter = (unsigned*)(ws + OFF_CTR);

  // ---- stage this WG's G slice into LDS ----
#if HAVE_TDM
  if (use_tdm) {
    if (wave == 0) {
      // D# group0: count=1 | lds_addr | 57-bit global addr | type=2
      unsigned lds_off = (unsigned)(unsigned long long)(const void*)Gs;
      unsigned long long ga =
          (unsigned long long)(ws + OFF_G + (size_t)rowbase * N_NODES);
      u32x4 g0;
      g0.x = 1u;                                   // count=1, user mode
      g0.y = lds_off;
      g0.z = (unsigned)(ga & 0xFFFFFFFFull);
      g0.w = (unsigned)((ga >> 32) & 0x1FFFFFFull) | (2u << 30);  // type=2
      // D# group1: data_size=4B, pad_enable, pad_interval=256DW, pad_amount=2DW
      i32x8 g1;
      g1[0] = (int)((2u << 16) | (1u << 20) | (7u << 22) | (1u << 25));
      g1[1] = (int)(512u << 16);   // abar_addr=0 | tensor_dim0[15:0]=512
      g1[2] = (int)(512u << 16);   // tensor_dim0[31:16]=0 | tensor_dim1[15:0]=512
      g1[3] = (int)(512u << 16);   // tensor_dim1[31:16]=0 | tile_dim0=512
      g1[4] = 32;                  // tile_dim1=32 | tile_dim2=0
      g1[5] = 512;                 // tensor_dim0_stride[31:0]=512
      g1[6] = 0;
      g1[7] = 0;                   // tensor_dim1_stride=0 (2D tile)
      i32x4 g2 = {0, 0, 0, 0};
      i32x4 g3 = {0, 0, 0, 0};
#if defined(__clang_major__) && __clang_major__ >= 23
      i32x8 g4 = {0, 0, 0, 0, 0, 0, 0, 0};
      __builtin_amdgcn_tensor_load_to_lds(g0, g1, g2, g3, g4, 0);
#else
      __builtin_amdgcn_tensor_load_to_lds(g0, g1, g2, g3, 0);
#endif
      __builtin_amdgcn_s_wait_tensorcnt(0);
    }
  } else
#endif
  {
    (void)use_tdm;
    for (int idx = tid; idx < ROWS_WG * N_NODES; idx += TPB) {
      int r = idx >> 9, i = idx & 511;
      int k = rowbase + r;
      float v = 0.0f;
      if (i != k) { float z = zval(tensions, i, k); v = (i < k) ? z : -z; }
      Gs[r * GPITCH + i + ((i >> 8) << 1)] = v;    // +2 pad past column 255
    }
  }
  for (int kk = tid; kk < N_NODES; kk += TPB) u_lds[kk * 4 + 3] = 0.0f;

  // wave/lane roles
  const int tau  = wave & 1;          // 16-row D tile
  const int kq   = wave >> 1;         // 128-wide K slice
  const int hi   = lane >> 4;
  const int ncol = lane & 15;
  const int bidx = (ncol < 3) ? ncol : 3;
  const float* grow = &Gs[(tau * 16 + (lane & 15)) * GPITCH + ((kq >= 2) ? 2 : 0)];

  const bool updater = (wave < 2) && (ncol < 3);
  float csv[8], imv[8], cmv[8], mmv[8], vel[8];
  if (updater) {
#pragma unroll
    for (int v = 0; v < 8; ++v) {
      int k = rowbase + tau * 16 + v + 8 * hi;
      csv[v] = ws[OFF_CS + k]; imv[v] = ws[OFF_IM + k];
      cmv[v] = ws[OFF_CM + k]; mmv[v] = ws[OFF_MM + k];
      vel[v] = 0.0f;
    }
  }

  for (int t = 0; t < N_FRAMES; ++t) {
    const float* uin  = (t & 1) ? u1 : u0;
    float*       uout = (t & 1) ? u0 : u1;

    // phase 1: stage full u (512x3) into LDS (async direct-to-LDS when available)
#if HAVE_ASYNC_LDS
#pragma unroll
    for (int e = 0; e < 6; ++e) {
      int idx = tid + e * TPB;                 // 0..1535
      int k = idx / 3, j = idx - 3 * k;
      __builtin_amdgcn_global_load_async_to_lds_b32(
          ASP_GLOBAL(uin + idx), ASP_LDS(u_lds + k * 4 + j), 0, 0);
    }
    __builtin_amdgcn_s_wait_asynccnt(0);
#else
    for (int idx = tid; idx < N_NODES * 3; idx += TPB) {
      int k = idx / 3, j = idx - 3 * k;
      u_lds[k * 4 + j] = uin[idx];
    }
#endif
    __syncthreads();

    // phase 2: y-tile partial = G_tile(16x128) * u(128x3) via fp32 WMMA
    v8f a0 = {0,0,0,0,0,0,0,0}, a1 = a0, a2 = a0, a3 = a0;
    const int kbase = kq * 128;
#pragma unroll
    for (int cc = 0; cc < 8; ++cc) {
      int k0 = kbase + cc * 16;
      a0 = wmma_step(grow, u_lds, k0 + 0,  hi, bidx, a0);
      a1 = wmma_step(grow, u_lds, k0 + 4,  hi, bidx, a1);
      a2 = wmma_step(grow, u_lds, k0 + 8,  hi, bidx, a2);
      a3 = wmma_step(grow, u_lds, k0 + 12, hi, bidx, a3);
    }
    part[wave * 32 + lane] = (a0 + a1) + (a2 + a3);
    __syncthreads();

    // phase 3: reduce K slices; waves 0/1 own tiles 0/1 and update state
    if (wave < 2) {
      v8f y = part[(tau + 0) * 32 + lane] + part[(tau + 2) * 32 + lane]
            + part[(tau + 4) * 32 + lane] + part[(tau + 6) * 32 + lane];
      if (ncol < 3) {
        const int j = ncol;
        const bool havef = ((t & 15) == 0);
        const float* fr = forces + (size_t)(t >> 4) * (N_NODES * 3);
        float recp = 0.0f;
#pragma unroll
        for (int v = 0; v < 8; ++v) {
          int k = rowbase + tau * 16 + v + 8 * hi;   // D row = vgpr + 8*hi
          float uo  = u_lds[k * 4 + j];
          float f   = havef ? fr[k * 3 + j] : 0.0f;
          float acc = f + (uo * csv[v] - y[v]) * imv[v];
          float v1  = vel[v] + acc;
          vel[v]    = v1 * 0.98f;                    // DAMPING
          uout[k * 3 + j] = uo + v1 * cmv[v];
          if (j == 0) recp += mmv[v] * acc;
        }
        if (j == 0) recs[tau * 2 + hi] = recp;
      }
    }
    __syncthreads();
    if (tid == 0)
      ws[OFF_RECP + t * NWG + wg] = recs[0] + recs[1] + recs[2] + recs[3];

    // grid barrier: release stores, count, acquire
    __threadfence();
    __syncthreads();
    if (tid == 0) {
      __hip_atomic_fetch_add(counter, 1u, __ATOMIC_ACQ_REL, __HIP_MEMORY_SCOPE_AGENT);
      unsigned target = (unsigned)(t + 1) * NWG;
      while (__hip_atomic_load(counter, __ATOMIC_RELAXED, __HIP_MEMORY_SCOPE_AGENT) < target)
        __builtin_amdgcn_s_sleep(2);
    }
    __syncthreads();
    __builtin_amdgcn_fence(__ATOMIC_ACQUIRE, "agent");
  }
}

// ---------------- tail: DFT + band-limited 256x upsample --------------------
__global__ void reduce_rec(float* __restrict__ ws) {
  int t = blockIdx.x * blockDim.x + threadIdx.x;
  if (t < N_FRAMES) {
    float s = 0.0f;
    for (int g = 0; g < NWG; ++g) s += ws[OFF_RECP + t * NWG + g];
    ws[OFF_REC + t] = s;
  }
}

__global__ void dft_kernel(float* __restrict__ ws) {
  int k = blockIdx.x * blockDim.x + threadIdx.x;
  if (k >= N_COEF) return;
  float re = 0.0f, im = 0.0f;
  for (int t = 0; t < N_FRAMES; ++t) {
    unsigned ph = ((unsigned)(k * t)) & 2047u;       // exact phase reduction
    float s, c; __sincosf((float)ph * (TWO_PI / 2048.0f), &s, &c);
    float r = ws[OFF_REC + t];
    re += r * c; im -= r * s;
  }
  ws[OFF_XR + k] = re;
  ws[OFF_XI + k] = im;
}

__global__ void synth_kernel(const float* __restrict__ ws, float* __restrict__ out) {
  __shared__ float sXr[N_COEF], sXi[N_COEF];
  for (int i = threadIdx.x; i < N_COEF; i += TPB) {
    sXr[i] = ws[OFF_XR + i];
    sXi[i] = ws[OFF_XI + i];
  }
  __syncthreads();
  const unsigned n = blockIdx.x * TPB + threadIdx.x;
  const float w = TWO_PI / (float)N_SAMP;
  float dsn, dc; __sincosf((float)n * w, &dsn, &dc);
  float sum = sXr[0];
  for (int g = 0; g < 16; ++g) {                     // bins 1..1024
    int kb = 1 + g * 64;
    unsigned ph = ((unsigned)kb * n) & (N_SAMP - 1u);
    float s, c; __sincosf((float)ph * w, &s, &c);
#pragma unroll 8
    for (int i = 0; i < 64; ++i) {
      int k = kb + i;
      sum += 2.0f * (sXr[k] * c - sXi[k] * s);
      float cn = c * dc - s * dsn;
      s = s * dc + c * dsn;
      c = cn;
    }
  }
  out[n] = sum * (1.0f / 2048.0f);
}

// ---------------- entry ------------------------------------------------------
extern "C" void kernel_launch(void* const* d_in, const int* in_sizes, int n_in,
                              void* d_out, int out_size, void* d_ws, size_t ws_size,
                              hipStream_t stream) {
  (void)in_sizes; (void)n_in; (void)out_size;
  // setup_inputs order: nodes, masses, tensions, mixer, constrained_mask, forces
  const float* masses   = (const float*)d_in[1];
  const float* tensions = (const float*)d_in[2];
  const float* mixer    = (const float*)d_in[3];
  const float* cmask    = (const float*)d_in[4];
  const float* forces   = (const float*)d_in[5];
  float* ws  = (float*)d_ws;
  float* out = (float*)d_out;
  if (ws_size < WS_BASE_FLOATS * sizeof(float)) return;

  const int use_tdm = (ws_size >= (size_t)WS_TDM_FLOATS * sizeof(float)) ? 1 : 0;

  prep_kernel<<<1, N_NODES, 0, stream>>>(masses, tensions, mixer, cmask, ws);
  if (use_tdm)
    prep_g<<<(N_NODES * N_NODES) / TPB, TPB, 0, stream>>>(tensions, ws);
  sim_kernel<<<NWG, TPB, 0, stream>>>(tensions, forces, ws, use_tdm);
  reduce_rec<<<N_FRAMES / TPB, TPB, 0, stream>>>(ws);
  dft_kernel<<<(N_COEF + TPB - 1) / TPB, TPB, 0, stream>>>(ws);
  synth_kernel<<<N_SAMP / TPB, TPB, 0, stream>>>(ws, out);
}